// REM_39316130627787
// MI455X (gfx1250) — compile-verified
//
#include <hip/hip_runtime.h>
#include <cstdint>

// REM positional-encoding generator for MI455X (gfx1250).
//
// The entire (Q, K, 8) fp32 output depends only on t = |i-j| (0 if > 200) in
// [0,200], so we build a 201x8 LUT in LDS once per workgroup and then the
// kernel is pure store bandwidth (134 MB @ 23.3 TB/s ~ 5.8 us floor).
// Stores use the CDNA5 async LDS->global path
// (global_store_async_from_lds_b128, ASYNCcnt-tracked): each lane moves
// 16 B directly from the LDS LUT row to global memory with no VGPR
// round-trip; one wave instruction streams 512 coalesced bytes.

#define TRUNC_T 200
#define NH 8

__global__ __launch_bounds__(256) void rem_lut_stream_kernel(
    const float* __restrict__ eta_p,
    const float* __restrict__ nu_p,
    const float* __restrict__ theta_p,
    const int*   __restrict__ qlen_p,
    const int*   __restrict__ klen_p,
    float* __restrict__ out)
{
    __shared__ float lut[(TRUNC_T + 1) * NH];   // 201 * 32 B = 6432 B

    const int   qlen  = *qlen_p;
    const int   klen  = *klen_p;
    const float eta   = *eta_p;
    const float nu    = *nu_p;
    const float theta = *theta_p;

    const float lam = tanhf(eta);
    const float gam = 1.0f / (1.0f + expf(-nu));

    // ---- Build the 201-entry LUT (8 floats per t) ----
    for (int t = threadIdx.x; t <= TRUNC_T; t += blockDim.x) {
        const float T  = (float)t;
        const float gT = powf(gam, T);
        const float c  = cosf(theta * T);
        const float s  = sinf(theta * T);
        const float h0 = gT * c;               // heads 0,1: gamma^T * cos
        const float h2 = gT * s;               // heads 2,3: gamma^T * sin
        // Dilated heads: exponent is 0 when t not divisible -> value 1 (or sin(0)=0).
        const float h4 = (t % 2  == 0) ? powf(lam, T * 0.5f)   : 1.0f;
        const float h5 = (t % 4  == 0) ? powf(lam, T * 0.25f)  : 1.0f;
        float h6 = 1.0f, h7 = 0.0f;
        if (t % 8 == 0)  { const float Ld = T * 0.125f;
                           h6 = powf(gam, Ld) * cosf(theta * Ld); }
        if (t % 16 == 0) { const float Ld = T * 0.0625f;
                           h7 = powf(gam, Ld) * sinf(theta * Ld); }
        float* row = &lut[t * NH];
        row[0] = h0; row[1] = h0;
        row[2] = h2; row[3] = h2;
        row[4] = h4; row[5] = h5;
        row[6] = h6; row[7] = h7;
    }
    __syncthreads();

    // ---- Stream rows: async LDS->global b128 stores ----
    const int chunksPerRow = klen * 2;         // 16-byte chunks per output row
    for (int i = blockIdx.x; i < qlen; i += gridDim.x) {
        // Uniform per-row global base -> SGPR pair for the GVS-form store.
        const unsigned long long rowBase =
            (unsigned long long)(uintptr_t)out +
            (unsigned long long)i * (unsigned long long)klen * (NH * 4ull);

        for (int c = threadIdx.x; c < chunksPerRow; c += blockDim.x) {
            const int j = c >> 1;
            int t = i - j; if (t < 0) t = -t;
            if (t > TRUNC_T) t = 0;            // truncation maps to the t=0 row

            const unsigned ldsAddr =
                (unsigned)(uintptr_t)(&lut[t * NH]) + (unsigned)((c & 1) << 4);
            const unsigned gOff = (unsigned)c << 4;   // c * 16 bytes into the row

            // GVS: the global address is the uniform 64-bit base in saddr plus each lane's 32-bit offset in vaddr;
            // data comes straight from the LDS address in ldsAddr, 16 B per lane, counted by ASYNCcnt.
            asm volatile(
                "global_store_async_from_lds_b128 %0, %1, %2"
                :
                : "v"(gOff), "v"(ldsAddr), "s"(rowBase)
                : "memory");
        }
    }

    // Drain outstanding async LDS->global transfers before wave exit.
    asm volatile("s_wait_asynccnt 0" ::: "memory");
}

extern "C" void kernel_launch(void* const* d_in, const int* in_sizes, int n_in,
                              void* d_out, int out_size, void* d_ws, size_t ws_size,
                              hipStream_t stream) {
    (void)in_sizes; (void)n_in; (void)out_size; (void)d_ws; (void)ws_size;

    const float* eta   = (const float*)d_in[0];
    const float* nu    = (const float*)d_in[1];
    const float* theta = (const float*)d_in[2];
    const int*   qlen  = (const int*)d_in[3];
    const int*   klen  = (const int*)d_in[4];

    // qlen/klen values live on device; use a fixed grid + grid-stride rows.
    dim3 grid(2048), block(256);
    rem_lut_stream_kernel<<<grid, block, 0, stream>>>(
        eta, nu, theta, qlen, klen, (float*)d_out);
}